// Speller_57775900066437
// MI455X (gfx1250) — compile-verified
//
#include <hip/hip_runtime.h>
#include <hip/hip_bf16.h>

#define T_LEN 2048
#define BATCH 32
#define CDIM  128
#define HDIM  256
#define VDIM  40
#define STEPS 256
#define KX    (VDIM + CDIM)           // 168
#define PLANE (3*BATCH*HDIM)          // 24576 floats per h/c snapshot
#define RAWSZ (BATCH*STEPS*VDIM)      // 327680 floats per raw copy

typedef __attribute__((ext_vector_type(2))) float v2f;
typedef __attribute__((ext_vector_type(8))) float v8f;

__device__ __forceinline__ float sigf(float x) { return 1.0f / (1.0f + __expf(-x)); }

// wave-uniform wave id (forces SALU loop control; keeps EXEC all-ones for WMMA)
__device__ __forceinline__ int wave_id() {
  return __builtin_amdgcn_readfirstlane((int)(threadIdx.x >> 5));
}

// OUT(16x16) += A(16xK) * W(16xK)^T, W stored row-major [N][K] (weight rows).
__device__ __forceinline__ v8f wmma_acc(const float* A, int lda,
                                        const float* W, int ldw,
                                        int K, v8f acc) {
  const int lane = threadIdx.x & 31;
  const int hf   = lane >> 4;
  const int l15  = lane & 15;
  const float* ar = A + l15 * lda + 2 * hf;
  const float* wr = W + l15 * ldw + 2 * hf;
#pragma unroll 4
  for (int k = 0; k < K; k += 4) {
    v2f av, bv;
    av.x = ar[k]; av.y = ar[k + 1];
    bv.x = wr[k]; bv.y = wr[k + 1];
    acc = __builtin_amdgcn_wmma_f32_16x16x4_f32(false, av, false, bv,
                                                (short)0, acc, false, false);
  }
  return acc;
}

// OUT(16x16) += A(16xK) * B(KxN tile), B stored row-major K-major [K][ldb]
// (used for q = h2 @ Wq where Wq is [H][C]). Lane-contiguous B loads.
__device__ __forceinline__ v8f wmma_acc_kmajor(const float* A, int lda,
                                               const float* B, int ldb,
                                               int K, v8f acc) {
  const int lane = threadIdx.x & 31;
  const int hf   = lane >> 4;
  const int l15  = lane & 15;
  const float* ar = A + l15 * lda + 2 * hf;
  const float* br = B + l15;                 // column n0+l15
#pragma unroll 4
  for (int k = 0; k < K; k += 4) {
    v2f av, bv;
    av.x = ar[k];                  av.y = ar[k + 1];
    bv.x = br[(k + 2 * hf) * ldb]; bv.y = br[(k + 2 * hf + 1) * ldb];
    acc = __builtin_amdgcn_wmma_f32_16x16x4_f32(false, av, false, bv,
                                                (short)0, acc, false, false);
  }
  return acc;
}

// ---------------- init: h/c state ping-pong buffers, zero <sos> word ----------
__global__ void k_init(const float* __restrict__ h0, const float* __restrict__ c0,
                       float* __restrict__ hbuf, float* __restrict__ cbuf,
                       float* __restrict__ word) {
  int i = blockIdx.x * blockDim.x + threadIdx.x;
  if (i < PLANE) { hbuf[i] = h0[i]; cbuf[i] = c0[i]; }
  if (i < BATCH * VDIM) word[i] = 0.0f;
}

// ---------------- query: q[32,128] = h2[32,256] @ Wq[256,128] + bq, via WMMA --
__global__ void k_query(const float* __restrict__ h2, const float* __restrict__ Wq,
                        const float* __restrict__ bq, float* __restrict__ q_ws) {
  const int wid  = wave_id();
  const int lane = threadIdx.x & 31;
  const int hf = lane >> 4, l15 = lane & 15;
  __builtin_prefetch(Wq + wid * (HDIM * CDIM / 8), 0, 1);
#pragma unroll
  for (int tile = wid; tile < 16; tile += 8) {       // 2 M-tiles x 8 N-tiles
    const int mt = tile & 1, nt = tile >> 1;
    const int m0 = mt * 16, n0 = nt * 16;
    v8f acc = {};
    acc = wmma_acc_kmajor(h2 + m0 * HDIM, HDIM, Wq + n0, CDIM, HDIM, acc);
    const int n = n0 + l15;
    const float bn = bq[n];
#pragma unroll
    for (int r = 0; r < 8; ++r)
      q_ws[(m0 + r + 8 * hf) * CDIM + n] = acc[r] + bn;
  }
}

// ---------------- attention: energy, masked softmax, ctx, x=[word|ctx] -------
__global__ void k_attention(const float* __restrict__ ls, const int* __restrict__ lens,
                            const float* __restrict__ q_ws, const float* __restrict__ word,
                            float* __restrict__ ctx_ws, float* __restrict__ x_ws,
                            float* __restrict__ atts, int step) {
  __shared__ float sh_q[CDIM];
  __shared__ float sh_e[T_LEN];
  __shared__ float sh_red[256];
  __shared__ float sh_pctx[2][CDIM];
  const int b   = blockIdx.x;
  const int tid = threadIdx.x;
  const int len = lens[b];

  if (tid < CDIM) sh_q[tid] = q_ws[b * CDIM + tid];
  __syncthreads();

  // energies: one wave per t, lanes split C (coalesced 512B rows, L2-resident)
  const int wid = tid >> 5, lane = tid & 31;
  for (int t = wid; t < T_LEN; t += 8) {
    const float* row = ls + ((size_t)t * BATCH + b) * CDIM;
    float s = 0.0f;
#pragma unroll
    for (int j = 0; j < 4; ++j) { int c = lane + 32 * j; s += row[c] * sh_q[c]; }
    for (int off = 16; off > 0; off >>= 1) s += __shfl_xor(s, off, 32);
    if (lane == 0) sh_e[t] = s;
  }
  __syncthreads();

  // masked softmax: block max
  float lmax = -3.0e38f;
  for (int t = tid; t < T_LEN; t += 256) if (t < len) lmax = fmaxf(lmax, sh_e[t]);
  sh_red[tid] = lmax; __syncthreads();
  for (int off = 128; off > 0; off >>= 1) {
    if (tid < off) sh_red[tid] = fmaxf(sh_red[tid], sh_red[tid + off]);
    __syncthreads();
  }
  const float mx = sh_red[0]; __syncthreads();

  float lsum = 0.0f;
  for (int t = tid; t < T_LEN; t += 256) {
    float v = (t < len) ? __expf(sh_e[t] - mx) : 0.0f;
    sh_e[t] = v; lsum += v;
  }
  sh_red[tid] = lsum; __syncthreads();
  for (int off = 128; off > 0; off >>= 1) {
    if (tid < off) sh_red[tid] += sh_red[tid + off];
    __syncthreads();
  }
  const float inv = 1.0f / sh_red[0]; __syncthreads();

  for (int t = tid; t < T_LEN; t += 256) {
    float a = sh_e[t] * inv;
    sh_e[t] = a;
    atts[((size_t)step * BATCH + b) * T_LEN + t] = a;   // atts[step][b][t]
  }
  __syncthreads();

  // ctx[c] = sum_t att[t]*ls[t][b][c]; 2 half-T partials x 128 channels
  const int c  = tid & (CDIM - 1);
  const int hb = tid >> 7;
  float s = 0.0f;
  for (int t = hb * (T_LEN / 2); t < (hb + 1) * (T_LEN / 2); ++t)
    s += sh_e[t] * ls[((size_t)t * BATCH + b) * CDIM + c];
  sh_pctx[hb][c] = s;
  __syncthreads();
  if (tid < CDIM) {
    float cv = sh_pctx[0][tid] + sh_pctx[1][tid];
    ctx_ws[b * CDIM + tid] = cv;
    x_ws[b * KX + VDIM + tid] = cv;
  }
  if (tid < VDIM) x_ws[b * KX + tid] = word[b * VDIM + tid];
}

// ---------------- one LSTM cell: each block owns 16 H-cols of all 4 gates ----
// 16 blocks x 8 waves, exactly 1 WMMA tile per wave -> minimal critical path.
__global__ void k_lstm(const float* __restrict__ xin, int K,
                       const float* __restrict__ Wih, const float* __restrict__ Whh,
                       const float* __restrict__ bih, const float* __restrict__ bhh,
                       const float* __restrict__ hprev, const float* __restrict__ cprev,
                       float* __restrict__ hnew, float* __restrict__ cnew) {
  __shared__ float gl[BATCH][4 * 16];   // [b][sec*16 + j] raw gate pre-activations
  const int hbase = blockIdx.x * 16;
  const int wid  = wave_id();
  const int lane = threadIdx.x & 31;
  const int hf = lane >> 4, l15 = lane & 15;

  // 8 tiles: sec(4) x mt(2); 8 waves -> 1 tile each (uniform per wave)
  {
    const int mt  = wid & 1;
    const int sec = wid >> 1;
    const int n0  = sec * HDIM + hbase;             // gate column in [0,1024)
    const int m0  = mt * 16;
    __builtin_prefetch(Wih + n0 * K, 0, 1);
    __builtin_prefetch(Whh + n0 * HDIM, 0, 1);
    v8f acc = {};
    acc = wmma_acc(xin   + m0 * K,    K,    Wih + n0 * K,    K,    K,    acc);
    acc = wmma_acc(hprev + m0 * HDIM, HDIM, Whh + n0 * HDIM, HDIM, HDIM, acc);
    const int cl = sec * 16 + l15;
#pragma unroll
    for (int r = 0; r < 8; ++r) gl[m0 + r + 8 * hf][cl] = acc[r];
  }
  __syncthreads();

  // pointwise: i,f,g,o all in-block; 512 elems over 256 threads
  for (int e = threadIdx.x; e < BATCH * 16; e += 256) {
    const int b = e >> 4, j = e & 15;
    const int gc = hbase + j;
    float iv = gl[b][j]        + bih[gc]            + bhh[gc];
    float fv = gl[b][16 + j]   + bih[HDIM + gc]     + bhh[HDIM + gc];
    float gv = gl[b][32 + j]   + bih[2 * HDIM + gc] + bhh[2 * HDIM + gc];
    float ov = gl[b][48 + j]   + bih[3 * HDIM + gc] + bhh[3 * HDIM + gc];
    float co = cprev[b * HDIM + gc];
    float cn = sigf(fv) * co + sigf(iv) * tanhf(gv);
    float hn = sigf(ov) * tanhf(cn);
    cnew[b * HDIM + gc] = cn;
    hnew[b * HDIM + gc] = hn;
  }
}

// ---------------- outfc (WMMA + leaky) then [32,40] projection ----------------
__global__ void k_outfc(const float* __restrict__ h2n, const float* __restrict__ ctx,
                        const float* __restrict__ fcW, const float* __restrict__ fcb,
                        const float* __restrict__ oW, const float* __restrict__ ob,
                        float* __restrict__ word, float* __restrict__ raw, int step) {
  __shared__ float sh_cat[BATCH][HDIM + CDIM];   // 32 x 384
  __shared__ float sh_so[BATCH][HDIM];           // 32 x 256
  const int tid = threadIdx.x;                   // 512 threads / 16 waves
  for (int i = tid; i < BATCH * (HDIM + CDIM); i += 512) {
    int b = i / (HDIM + CDIM), k = i % (HDIM + CDIM);
    sh_cat[b][k] = (k < HDIM) ? h2n[b * HDIM + k] : ctx[b * CDIM + (k - HDIM)];
  }
  __syncthreads();

  const int wid = wave_id();
  const int lane = tid & 31;
  const int hf = lane >> 4, l15 = lane & 15;
  __builtin_prefetch(fcW + wid * (HDIM * (HDIM + CDIM) / 16), 0, 1);
#pragma unroll
  for (int tile = wid; tile < 32; tile += 16) {          // 2 M-tiles x 16 N-tiles
    const int mt = tile & 1, nt = tile >> 1;
    const int m0 = mt * 16, n0 = nt * 16;
    v8f acc = {};
    acc = wmma_acc(&sh_cat[m0][0], HDIM + CDIM,
                   fcW + n0 * (HDIM + CDIM), HDIM + CDIM, HDIM + CDIM, acc);
    const int n = n0 + l15;
    const float bn = fcb[n];
#pragma unroll
    for (int r = 0; r < 8; ++r) {                        // fused bias + leaky(0.2)
      float v = acc[r] + bn;
      sh_so[m0 + r + 8 * hf][n] = (v > 0.0f) ? v : 0.2f * v;
    }
  }
  __syncthreads();

  for (int i = tid; i < BATCH * VDIM; i += 512) {
    const int b = i / VDIM, v = i % VDIM;
    float s = ob[v];
    const float* wrow = oW + v * HDIM;
    for (int k = 0; k < HDIM; ++k) s += sh_so[b][k] * wrow[k];
    word[b * VDIM + v] = s;                              // next-step input
    const size_t o = ((size_t)b * STEPS + step) * VDIM + v;
    raw[o] = s;                                          // output copy 1
    raw[RAWSZ + o] = s;                                  // output copy 2
  }
}

extern "C" void kernel_launch(void* const* d_in, const int* in_sizes, int n_in,
                              void* d_out, int out_size, void* d_ws, size_t ws_size,
                              hipStream_t stream) {
  const float* ls   = (const float*)d_in[0];
  const int*   lens = (const int*)  d_in[1];
  const float* h0   = (const float*)d_in[2];
  const float* c0   = (const float*)d_in[3];
  const float* Wq   = (const float*)d_in[4];
  const float* bq   = (const float*)d_in[5];
  const float* aWih = (const float*)d_in[6];
  const float* aWhh = (const float*)d_in[7];
  const float* abih = (const float*)d_in[8];
  const float* abhh = (const float*)d_in[9];
  const float* w1ih = (const float*)d_in[10];
  const float* w1hh = (const float*)d_in[11];
  const float* b1ih = (const float*)d_in[12];
  const float* b1hh = (const float*)d_in[13];
  const float* w2ih = (const float*)d_in[14];
  const float* w2hh = (const float*)d_in[15];
  const float* b2ih = (const float*)d_in[16];
  const float* b2hh = (const float*)d_in[17];
  const float* fcW  = (const float*)d_in[18];
  const float* fcb  = (const float*)d_in[19];
  const float* oW   = (const float*)d_in[20];
  const float* ob   = (const float*)d_in[21];

  float* ws     = (float*)d_ws;
  float* x_ws   = ws;                        // [32][168]
  float* ctx_ws = x_ws + BATCH * KX;         // [32][128]
  float* q_ws   = ctx_ws + BATCH * CDIM;     // [32][128]
  float* word   = q_ws + BATCH * CDIM;       // [32][40]
  float* hbuf   = word + BATCH * VDIM;       // [2][3][32][256] ping-pong
  float* cbuf   = hbuf + 2 * PLANE;          // [2][3][32][256]
  float* out    = (float*)d_out;
  float* atts   = out + 2 * RAWSZ;           // [256][32][2048]

  k_init<<<(PLANE + 255) / 256, 256, 0, stream>>>(h0, c0, hbuf, cbuf, word);

  for (int step = 0; step < STEPS; ++step) {
    const int p = step & 1;
    const float* hp = hbuf + p * PLANE;
    const float* cp = cbuf + p * PLANE;
    float* hn = hbuf + (1 - p) * PLANE;
    float* cn = cbuf + (1 - p) * PLANE;

    k_query<<<1, 256, 0, stream>>>(hp + 2 * BATCH * HDIM, Wq, bq, q_ws);
    k_attention<<<BATCH, 256, 0, stream>>>(ls, lens, q_ws, word,
                                           ctx_ws, x_ws, atts, step);
    k_lstm<<<16, 256, 0, stream>>>(x_ws, KX, aWih, aWhh, abih, abhh,
                                   hp, cp, hn, cn);
    k_lstm<<<16, 256, 0, stream>>>(hn, HDIM, w1ih, w1hh, b1ih, b1hh,
                                   hp + BATCH * HDIM, cp + BATCH * HDIM,
                                   hn + BATCH * HDIM, cn + BATCH * HDIM);
    k_lstm<<<16, 256, 0, stream>>>(hn + BATCH * HDIM, HDIM, w2ih, w2hh, b2ih, b2hh,
                                   hp + 2 * BATCH * HDIM, cp + 2 * BATCH * HDIM,
                                   hn + 2 * BATCH * HDIM, cn + 2 * BATCH * HDIM);
    k_outfc<<<1, 512, 0, stream>>>(hn + 2 * BATCH * HDIM, ctx_ws, fcW, fcb, oW, ob,
                                   word, out, step);
  }
}